// STFNNWithMeteo_85572928406046
// MI455X (gfx1250) — compile-verified
//
#include <hip/hip_runtime.h>
#include <math.h>

typedef __attribute__((ext_vector_type(2))) float v2f;
typedef __attribute__((ext_vector_type(8))) float v8f;

static constexpr int B_N = 8192;
static constexpr int K_N = 6;
static constexpr int T_N = 24;
static constexpr int F_N = 6;
static constexpr int H_N = 64;
static constexpr int M_N = 16;
static constexpr int NSEQ1 = B_N * K_N;  // 49152
static constexpr int NSEQ2 = B_N;        // 8192

__device__ __forceinline__ v8f wmma4(v2f a, v2f b, v8f c) {
  // D = A(16x4,f32) * B(4x16,f32) + C(16x16,f32)
  return __builtin_amdgcn_wmma_f32_16x16x4_f32(false, a, false, b, (short)0, c,
                                               false, false);
}

// ---------------------------------------------------------------------------
// Fused encoder (last-row only) + spatio-temporal code + WMMA MLP head.
// One thread = one sequence; each wave then does code(32x16) @ mlp_w(16x64).
// Grid must tile nseq exactly with 256-thread blocks (49152/8192 both do).
// ---------------------------------------------------------------------------
__global__ void __launch_bounds__(256)
encode_mlp_kernel(const float* __restrict__ coord,   // (nseq,3)
                  const float* __restrict__ meo,     // (nseq,T,F)
                  const float* __restrict__ attn_w,  // (4,F,F)
                  const float* __restrict__ attn_b,  // (4,F)
                  const float* __restrict__ ffn_w1,  // (F,H)
                  const float* __restrict__ ffn_b1,  // (H)
                  const float* __restrict__ ffn_w2,  // (H,F)
                  const float* __restrict__ ffn_b2,  // (F)
                  const float* __restrict__ ln,      // (4,F)
                  const float* __restrict__ mlp_w,   // (16,H)
                  const float* __restrict__ mlp_b,   // (H)
                  float* __restrict__ p_out)         // (nseq,H)
{
  __shared__ float codebuf[256][17];  // padded stride: conflict-free frag reads
  const int tid = threadIdx.x;
  const int seq = blockIdx.x * 256 + tid;

  const float* xrow = meo + (size_t)seq * (T_N * F_N);
  const float* wq = attn_w;
  const float* wk = attn_w + F_N * F_N;
  const float* wv = attn_w + 2 * F_N * F_N;
  const float* wo = attn_w + 3 * F_N * F_N;
  const float* bq = attn_b;
  const float* bk = attn_b + F_N;
  const float* bv = attn_b + 2 * F_N;
  const float* bo = attn_b + 3 * F_N;

  float x23[F_N];
#pragma unroll
  for (int d = 0; d < F_N; ++d) x23[d] = xrow[(T_N - 1) * F_N + d];

  float q23[F_N];
#pragma unroll
  for (int j = 0; j < F_N; ++j) {
    float a = bq[j];
#pragma unroll
    for (int d = 0; d < F_N; ++d) a += x23[d] * wq[d * F_N + j];
    q23[j] = a;
  }
  const float scale = rsqrtf((float)F_N);

  // Online-softmax attention for query row 23 only.
  float mmax = -3.0e38f, ssum = 0.f;
  float oacc[F_N];
#pragma unroll
  for (int j = 0; j < F_N; ++j) oacc[j] = 0.f;

  for (int u = 0; u < T_N; ++u) {
    float xu[F_N];
#pragma unroll
    for (int d = 0; d < F_N; ++d) xu[d] = xrow[u * F_N + d];
    float sc = 0.f;
    float vu[F_N];
#pragma unroll
    for (int j = 0; j < F_N; ++j) {
      float kk = bk[j], vv = bv[j];
#pragma unroll
      for (int d = 0; d < F_N; ++d) {
        kk += xu[d] * wk[d * F_N + j];
        vv += xu[d] * wv[d * F_N + j];
      }
      sc += q23[j] * kk;
      vu[j] = vv;
    }
    sc *= scale;
    const float nm = fmaxf(mmax, sc);
    const float corr = expf(mmax - nm);
    const float wgt = expf(sc - nm);
    ssum = ssum * corr + wgt;
#pragma unroll
    for (int j = 0; j < F_N; ++j) oacc[j] = oacc[j] * corr + wgt * vu[j];
    mmax = nm;
  }
  const float inv = 1.f / ssum;
#pragma unroll
  for (int j = 0; j < F_N; ++j) oacc[j] *= inv;

  float o23[F_N];
#pragma unroll
  for (int j = 0; j < F_N; ++j) {
    float a = bo[j];
#pragma unroll
    for (int d = 0; d < F_N; ++d) a += oacc[d] * wo[d * F_N + j];
    o23[j] = a;
  }

  // LayerNorm 1 (gamma=ln[0], beta=ln[1])
  float x1[F_N];
  {
    float t[F_N], mu = 0.f;
#pragma unroll
    for (int d = 0; d < F_N; ++d) { t[d] = x23[d] + o23[d]; mu += t[d]; }
    mu *= (1.f / F_N);
    float var = 0.f;
#pragma unroll
    for (int d = 0; d < F_N; ++d) { const float z = t[d] - mu; var += z * z; }
    var *= (1.f / F_N);
    const float rstd = rsqrtf(var + 1e-6f);
#pragma unroll
    for (int d = 0; d < F_N; ++d)
      x1[d] = (t[d] - mu) * rstd * ln[d] + ln[F_N + d];
  }

  // FFN on row 23
  float fv[F_N];
#pragma unroll
  for (int j = 0; j < F_N; ++j) fv[j] = ffn_b2[j];
  for (int h = 0; h < H_N; ++h) {
    float a = ffn_b1[h];
#pragma unroll
    for (int d = 0; d < F_N; ++d) a += x1[d] * ffn_w1[d * H_N + h];
    a = fmaxf(a, 0.f);
#pragma unroll
    for (int j = 0; j < F_N; ++j) fv[j] += a * ffn_w2[h * F_N + j];
  }

  // LayerNorm 2 (gamma=ln[2], beta=ln[3]) -> encoded meteo (F_N)
  float met[F_N];
  {
    float t[F_N], mu = 0.f;
#pragma unroll
    for (int d = 0; d < F_N; ++d) { t[d] = x1[d] + fv[d]; mu += t[d]; }
    mu *= (1.f / F_N);
    float var = 0.f;
#pragma unroll
    for (int d = 0; d < F_N; ++d) { const float z = t[d] - mu; var += z * z; }
    var *= (1.f / F_N);
    const float rstd = rsqrtf(var + 1e-6f);
#pragma unroll
    for (int d = 0; d < F_N; ++d)
      met[d] = (t[d] - mu) * rstd * ln[2 * F_N + d] + ln[3 * F_N + d];
  }

  // st_code (10 dims) + met (6 dims) -> 16-dim row staged in LDS
  {
    const float t0 = coord[(size_t)seq * 3 + 0];
    codebuf[tid][0] = coord[(size_t)seq * 3 + 1];
    codebuf[tid][1] = coord[(size_t)seq * 3 + 2];
    const float TWO_PI = 6.28318530717958647692f;
    const float per[4] = {1.f, 7.f, 30.5f, 365.f};
#pragma unroll
    for (int p = 0; p < 4; ++p) {
      const float ang = TWO_PI * t0 / (per[p] * 24.f);
      codebuf[tid][2 + 2 * p] = sinf(ang);
      codebuf[tid][2 + 2 * p + 1] = cosf(ang);
    }
#pragma unroll
    for (int d = 0; d < F_N; ++d) codebuf[tid][10 + d] = met[d];
  }
  __syncthreads();

  // Each wave: two 16-row tiles x (K=16 -> 4 wmma k-steps) x (N=64 -> 4 tiles)
  const int lane = tid & 31;
  const int wave = tid >> 5;
  const int half = lane >> 4;
  const int n = lane & 15;
#pragma unroll
  for (int tb = 0; tb < 2; ++tb) {
    const int rowbase = wave * 32 + tb * 16;
    v2f afr[4];
#pragma unroll
    for (int kb = 0; kb < 4; ++kb) {
      afr[kb].x = codebuf[rowbase + n][kb * 4 + 2 * half];
      afr[kb].y = codebuf[rowbase + n][kb * 4 + 2 * half + 1];
    }
#pragma unroll
    for (int nt = 0; nt < 4; ++nt) {
      v8f acc = {};
#pragma unroll
      for (int kb = 0; kb < 4; ++kb) {
        const int kr = kb * 4 + 2 * half;
        v2f bfr;
        bfr.x = mlp_w[kr * H_N + nt * 16 + n];
        bfr.y = mlp_w[(kr + 1) * H_N + nt * 16 + n];
        acc = wmma4(afr[kb], bfr, acc);
      }
      const float bias = mlp_b[nt * 16 + n];
#pragma unroll
      for (int j = 0; j < 8; ++j) {
        const int row = rowbase + j + 8 * half;
        const int oseq = blockIdx.x * 256 + row;
        const float v = acc[j] + bias;
        p_out[(size_t)oseq * H_N + nt * 16 + n] = fmaxf(v, 0.f);
      }
    }
  }
}

// ---------------------------------------------------------------------------
// Ring integration: one wave per (b,k) pair; the 16-row WMMA tile is exactly
// the M_STEPS=16 integration points. 9->64 (wmma K=12 padded), 64->64 (wmma
// K=64), then 64->3 folded into a per-lane dot with w3·direction + reduction.
// ---------------------------------------------------------------------------
__global__ void __launch_bounds__(256)
ring_kernel(const float* __restrict__ scr_coord,  // (B,K,3)
            const float* __restrict__ tar_coord,  // (B,3)
            const float* __restrict__ scr_meo,    // (B,K,T,F)
            const float* __restrict__ scr_y,      // (B,K)
            const float* __restrict__ w1, const float* __restrict__ b1,
            const float* __restrict__ w2, const float* __restrict__ b2,
            const float* __restrict__ w3, const float* __restrict__ b3,
            float* __restrict__ y_est)            // (B*K)
{
  __shared__ float rbuf[8][16][13];  // 16 rows x 12 cols (K padded), stride 13
  __shared__ float hbuf[8][16][65];  // 16 rows x 64 cols, stride 65
  const int tid = threadIdx.x;
  const int wave = tid >> 5;
  const int lane = tid & 31;
  const int half = lane >> 4;
  const int n = lane & 15;
  const int pair = blockIdx.x * 8 + wave;  // < B*K (grid tiles exactly)
  const int bidx = pair / K_N;

  const float s0 = scr_coord[(size_t)pair * 3 + 0];
  const float s1 = scr_coord[(size_t)pair * 3 + 1];
  const float s2 = scr_coord[(size_t)pair * 3 + 2];
  const float d0 = tar_coord[(size_t)bidx * 3 + 0] - s0;
  const float d1 = tar_coord[(size_t)bidx * 3 + 1] - s1;
  const float d2 = tar_coord[(size_t)bidx * 3 + 2] - s2;
  const float nrm = sqrtf(d0 * d0 + d1 * d1 + d2 * d2);
  const float innrm = 1.f / (nrm + 1e-8f);
  const float dir0 = d0 * innrm, dir1 = d1 * innrm, dir2 = d2 * innrm;
  const float st0 = d0 * (1.f / M_N), st1 = d1 * (1.f / M_N),
              st2 = d2 * (1.f / M_N);

  const float* mp = scr_meo + ((size_t)pair * T_N + (T_N - 1)) * F_N;

  if (lane < 16) {  // lane == integration step j
    const float jj = (float)lane;
    rbuf[wave][lane][0] = s0 + st0 * jj;
    rbuf[wave][lane][1] = s1 + st1 * jj;
    rbuf[wave][lane][2] = s2 + st2 * jj;
#pragma unroll
    for (int d = 0; d < F_N; ++d) rbuf[wave][lane][3 + d] = mp[d];
    rbuf[wave][lane][9] = 0.f;
    rbuf[wave][lane][10] = 0.f;
    rbuf[wave][lane][11] = 0.f;
  }
  asm volatile("" ::: "memory");  // keep DS issue order (in-order per wave)

  // GEMM1: rin(16x12) @ w1(12x64), rows 9..11 are zero padding
  v2f afr1[3];
#pragma unroll
  for (int kb = 0; kb < 3; ++kb) {
    afr1[kb].x = rbuf[wave][n][kb * 4 + 2 * half];
    afr1[kb].y = rbuf[wave][n][kb * 4 + 2 * half + 1];
  }
#pragma unroll
  for (int nt = 0; nt < 4; ++nt) {
    v8f acc = {};
#pragma unroll
    for (int kb = 0; kb < 3; ++kb) {
      const int kr = kb * 4 + 2 * half;
      v2f bfr;
      bfr.x = (kr < 9) ? w1[kr * H_N + nt * 16 + n] : 0.f;
      bfr.y = (kr + 1 < 9) ? w1[(kr + 1) * H_N + nt * 16 + n] : 0.f;
      acc = wmma4(afr1[kb], bfr, acc);
    }
    const float bb = b1[nt * 16 + n];
#pragma unroll
    for (int j = 0; j < 8; ++j) {
      const float v = acc[j] + bb;
      hbuf[wave][j + 8 * half][nt * 16 + n] = fmaxf(v, 0.f);
    }
  }
  asm volatile("" ::: "memory");

  // GEMM2: h1(16x64) @ w2(64x64); fold relu + w3·dir into accumulator drain
  v2f afr2[16];
#pragma unroll
  for (int kb = 0; kb < 16; ++kb) {
    afr2[kb].x = hbuf[wave][n][kb * 4 + 2 * half];
    afr2[kb].y = hbuf[wave][n][kb * 4 + 2 * half + 1];
  }
  float local = 0.f;
#pragma unroll
  for (int nt = 0; nt < 4; ++nt) {
    v8f acc = {};
#pragma unroll
    for (int kb = 0; kb < 16; ++kb) {
      const int kr = kb * 4 + 2 * half;
      v2f bfr;
      bfr.x = w2[kr * H_N + nt * 16 + n];
      bfr.y = w2[(kr + 1) * H_N + nt * 16 + n];
      acc = wmma4(afr2[kb], bfr, acc);
    }
    const int dcol = nt * 16 + n;
    const float wd = w3[dcol * 3 + 0] * dir0 + w3[dcol * 3 + 1] * dir1 +
                     w3[dcol * 3 + 2] * dir2;
    const float bb = b2[dcol];
    float s = 0.f;
#pragma unroll
    for (int j = 0; j < 8; ++j) s += fmaxf(acc[j] + bb, 0.f);
    local += s * wd;
  }
#pragma unroll
  for (int off = 16; off > 0; off >>= 1) local += __shfl_xor(local, off, 32);
  if (lane == 0) {
    const float bdot = b3[0] * dir0 + b3[1] * dir1 + b3[2] * dir2;
    y_est[pair] = scr_y[pair] + local + (float)M_N * bdot;
  }
}

// ---------------------------------------------------------------------------
// Aggregation: per-b softmax-weighted combine. One thread per b.
// ---------------------------------------------------------------------------
__global__ void __launch_bounds__(256)
agg_kernel(const float* __restrict__ p_src,  // (B*K,H)
           const float* __restrict__ p_tar,  // (B,H)
           const float* __restrict__ y_est,  // (B*K)
           const float* __restrict__ aw1, const float* __restrict__ ab1,
           const float* __restrict__ aw2, const float* __restrict__ ab2,
           float* __restrict__ out)          // (B)
{
  const int b = blockIdx.x * 256 + threadIdx.x;
  const float* pt = p_tar + (size_t)b * H_N;
  float logits[K_N];
  for (int k = 0; k < K_N; ++k) {
    const float* ps = p_src + ((size_t)b * K_N + k) * H_N;
    float a[K_N];
#pragma unroll
    for (int j = 0; j < K_N; ++j) a[j] = ab1[j];
    for (int d = 0; d < H_N; ++d) {
      const float s = ps[d];
#pragma unroll
      for (int j = 0; j < K_N; ++j) a[j] += s * aw1[d * K_N + j];
    }
    for (int d = 0; d < H_N; ++d) {
      const float s = pt[d];
#pragma unroll
      for (int j = 0; j < K_N; ++j) a[j] += s * aw1[(H_N + d) * K_N + j];
    }
    float lg = ab2[0];
#pragma unroll
    for (int j = 0; j < K_N; ++j) lg += fmaxf(a[j], 0.f) * aw2[j];
    logits[k] = lg;
  }
  float mx = logits[0];
#pragma unroll
  for (int k = 1; k < K_N; ++k) mx = fmaxf(mx, logits[k]);
  float den = 0.f, o = 0.f;
#pragma unroll
  for (int k = 0; k < K_N; ++k) {
    const float w = expf(logits[k] - mx);
    den += w;
    o += w * y_est[(size_t)b * K_N + k];
  }
  out[b] = o / den;
}

extern "C" void kernel_launch(void* const* d_in, const int* in_sizes, int n_in,
                              void* d_out, int out_size, void* d_ws,
                              size_t ws_size, hipStream_t stream) {
  (void)in_sizes; (void)n_in; (void)out_size; (void)ws_size;
  const float* c_scr_coord = (const float*)d_in[0];
  const float* c_scr_meo   = (const float*)d_in[1];
  const float* c_tar_coord = (const float*)d_in[2];
  const float* c_tar_meo   = (const float*)d_in[3];
  const float* c_scr_y     = (const float*)d_in[4];
  // e1: indices 5..13, e2: 14..22
  const float* e1_attn_w = (const float*)d_in[5];
  const float* e1_attn_b = (const float*)d_in[6];
  const float* e1_ffn_w1 = (const float*)d_in[7];
  const float* e1_ffn_b1 = (const float*)d_in[8];
  const float* e1_ffn_w2 = (const float*)d_in[9];
  const float* e1_ffn_b2 = (const float*)d_in[10];
  const float* e1_ln     = (const float*)d_in[11];
  const float* e1_mlp_w  = (const float*)d_in[12];
  const float* e1_mlp_b  = (const float*)d_in[13];
  const float* e2_attn_w = (const float*)d_in[14];
  const float* e2_attn_b = (const float*)d_in[15];
  const float* e2_ffn_w1 = (const float*)d_in[16];
  const float* e2_ffn_b1 = (const float*)d_in[17];
  const float* e2_ffn_w2 = (const float*)d_in[18];
  const float* e2_ffn_b2 = (const float*)d_in[19];
  const float* e2_ln     = (const float*)d_in[20];
  const float* e2_mlp_w  = (const float*)d_in[21];
  const float* e2_mlp_b  = (const float*)d_in[22];
  const float* ring_w1 = (const float*)d_in[23];
  const float* ring_b1 = (const float*)d_in[24];
  const float* ring_w2 = (const float*)d_in[25];
  const float* ring_b2 = (const float*)d_in[26];
  const float* ring_w3 = (const float*)d_in[27];
  const float* ring_b3 = (const float*)d_in[28];
  const float* agg_w1 = (const float*)d_in[29];
  const float* agg_b1 = (const float*)d_in[30];
  const float* agg_w2 = (const float*)d_in[31];
  const float* agg_b2 = (const float*)d_in[32];

  float* ws = (float*)d_ws;
  float* p_src = ws;                                  // NSEQ1*H
  float* p_tar = p_src + (size_t)NSEQ1 * H_N;         // NSEQ2*H
  float* y_est = p_tar + (size_t)NSEQ2 * H_N;         // NSEQ1
  float* out = (float*)d_out;                         // (B)

  encode_mlp_kernel<<<NSEQ1 / 256, 256, 0, stream>>>(
      c_scr_coord, c_scr_meo, e1_attn_w, e1_attn_b, e1_ffn_w1, e1_ffn_b1,
      e1_ffn_w2, e1_ffn_b2, e1_ln, e1_mlp_w, e1_mlp_b, p_src);

  encode_mlp_kernel<<<NSEQ2 / 256, 256, 0, stream>>>(
      c_tar_coord, c_tar_meo, e2_attn_w, e2_attn_b, e2_ffn_w1, e2_ffn_b1,
      e2_ffn_w2, e2_ffn_b2, e2_ln, e2_mlp_w, e2_mlp_b, p_tar);

  ring_kernel<<<NSEQ1 / 8, 256, 0, stream>>>(
      c_scr_coord, c_tar_coord, c_scr_meo, c_scr_y, ring_w1, ring_b1, ring_w2,
      ring_b2, ring_w3, ring_b3, y_est);

  agg_kernel<<<B_N / 256, 256, 0, stream>>>(p_src, p_tar, y_est, agg_w1,
                                            agg_b1, agg_w2, agg_b2, out);
}